// SparseTF_86509231276655
// MI455X (gfx1250) — compile-verified
//
#include <hip/hip_runtime.h>
#include <hip/hip_bf16.h>
#include <cstdint>
#include <cstddef>

#define INPUT_DIM 20000
#define UNITS     8000
#define BATCH     256
#define KSPLIT    10                      // 20000 / 10 = 2000 k per block
#define KSLAB     (INPUT_DIM / KSPLIT)    // 2000, multiple of 16

typedef float v2f __attribute__((ext_vector_type(2)));
typedef float v8f __attribute__((ext_vector_type(8)));

// ---------------- zero fill (float4 grid-stride) ----------------
__global__ void zero_f4_kernel(float4* __restrict__ p, long long n4) {
    long long i = (long long)blockIdx.x * blockDim.x + threadIdx.x;
    long long stride = (long long)gridDim.x * blockDim.x;
    float4 z = make_float4(0.f, 0.f, 0.f, 0.f);
    for (; i < n4; i += stride) p[i] = z;
}

// ---------------- scatter sparse values into dense K ----------------
__global__ void scatter_kernel(const int* __restrict__ ind,
                               const float* __restrict__ val,
                               float* __restrict__ Kd, int nnz) {
    int i = blockIdx.x * blockDim.x + threadIdx.x;
    if (i < nnz) {
        int r = ind[2 * i + 0];
        int c = ind[2 * i + 1];
        atomicAdd(Kd + (size_t)r * UNITS + c, val[i]);   // duplicates add
    }
}

// ---------------- WMMA f32 split-K GEMM (partials -> atomicAdd) ----------------
// Block: 256 threads (8 waves). Block tile: M=256 (full batch) x N=64, K-slab 2000.
// Wave w: M-half (w>>2)*128, N-stripe (w&3)*16. Per wave: 8 M-tiles of 16x16
// (8 x v8f accumulators = 64 VGPRs) -> register headroom for fragment ILP.
// Grid: 125 N-blocks x 10 K-slabs = 1250 blocks = 10000 waves (occupancy +
// latency hiding), while every K element is still read from HBM exactly once.
__global__ __launch_bounds__(256) void gemm_splitk_kernel(
    const float* __restrict__ A,     // BATCH x INPUT_DIM
    const float* __restrict__ Kd,    // INPUT_DIM x UNITS (dense)
    float* __restrict__ out) {       // BATCH x UNITS (zeroed; atomic accumulate)
    // row stride 20 floats: 16 data + 4 pad -> conflict-free lane stride,
    // 16B-aligned float4 staging stores, 8B-aligned float2 fragment loads.
    __shared__ float panel[BATCH * 20];

    const int tid     = threadIdx.x;
    const int lane    = tid & 31;
    const int wave    = tid >> 5;
    const int lane15  = lane & 15;
    const int halfSel = lane >> 4;        // 0 for lanes 0-15, 1 for 16-31
    const int off     = halfSel << 1;     // K-slice offset 0 or 2 (ISA A/B layout)

    const int m0   = (wave >> 2) * 128;               // M-half of this wave
    const int ncol = blockIdx.x * 64 + (wave & 3) * 16 + lane15;  // 8000%64==0
    const int kbase = blockIdx.y * KSLAB;

    v8f acc[8];
#pragma unroll
    for (int t = 0; t < 8; ++t) acc[t] = (v8f)(0.0f);

    for (int k0 = kbase; k0 < kbase + KSLAB; k0 += 16) {   // 2000 % 16 == 0
        // Stage A[0:256, k0:k0+16]: thread m loads its own row, 64B contiguous.
        const float4* src = (const float4*)(A + (size_t)tid * INPUT_DIM + k0);
        float4 r0 = src[0], r1 = src[1], r2 = src[2], r3 = src[3];
        float4* dst = (float4*)&panel[tid * 20];
        dst[0] = r0; dst[1] = r1; dst[2] = r2; dst[3] = r3;
        __syncthreads();

#pragma unroll
        for (int kk = 0; kk < 16; kk += 4) {
            // B fragment: Kd[k0+kk+off, ncol], Kd[k0+kk+off+1, ncol]
            size_t bbase = (size_t)(k0 + kk + off) * UNITS + ncol;
            v2f b;
            b.x = Kd[bbase];
            b.y = Kd[bbase + UNITS];
            // Batch all 8 A-fragment LDS loads ahead of the WMMAs.
            v2f a[8];
#pragma unroll
            for (int t = 0; t < 8; ++t) {
                const int m = m0 + t * 16 + lane15;
                a[t] = *(const v2f*)&panel[m * 20 + kk + off];  // ds_load b64
            }
#pragma unroll
            for (int t = 0; t < 8; ++t) {
                acc[t] = __builtin_amdgcn_wmma_f32_16x16x4_f32(
                    /*neg_a=*/false, a[t], /*neg_b=*/false, b,
                    /*c_mod=*/(short)0, acc[t],
                    /*reuse_a=*/false, /*reuse_b=*/false);
            }
        }
        __syncthreads();
    }

    // C/D layout: VGPR j holds M = 16t + j (lanes 0-15) / +8 (lanes 16-31).
    // Split-K partial -> atomic accumulate into out (L2-resident, 8 MB).
#pragma unroll
    for (int t = 0; t < 8; ++t) {
#pragma unroll
        for (int j = 0; j < 8; ++j) {
            int row = m0 + t * 16 + j + (halfSel << 3);
            atomicAdd(out + (size_t)row * UNITS + ncol, acc[t][j]);
        }
    }
}

// ---------------- bias + tanh epilogue ----------------
__global__ void bias_tanh_kernel(float* __restrict__ out,
                                 const float* __restrict__ bias, int n) {
    int i = blockIdx.x * blockDim.x + threadIdx.x;
    if (i < n) out[i] = tanhf(out[i] + bias[i % UNITS]);
}

// ---------------- fallback (workspace too small): atomic SpMM ----------------
__global__ void spmm_atomic_kernel(const float* __restrict__ in,
                                   const int* __restrict__ ind,
                                   const float* __restrict__ val,
                                   float* __restrict__ out, int nnz) {
    int i = blockIdx.x;
    if (i >= nnz) return;
    int r = ind[2 * i + 0];
    int c = ind[2 * i + 1];
    float v = val[i];
    int b = threadIdx.x;  // blockDim.x == BATCH
    atomicAdd(out + (size_t)b * UNITS + c, v * in[(size_t)b * INPUT_DIM + r]);
}

extern "C" void kernel_launch(void* const* d_in, const int* in_sizes, int n_in,
                              void* d_out, int out_size, void* d_ws, size_t ws_size,
                              hipStream_t stream) {
    const float* inputs = (const float*)d_in[0];   // (256, 20000) f32
    const int*   ind    = (const int*)d_in[1];     // (NNZ, 2) i32
    const float* val    = (const float*)d_in[2];   // (NNZ,) f32
    const float* bias   = (const float*)d_in[3];   // (8000,) f32
    float*       out    = (float*)d_out;           // (256, 8000) f32
    const int nnz = in_sizes[1] / 2;
    const int nout = BATCH * UNITS;

    const size_t kbytes = (size_t)INPUT_DIM * UNITS * sizeof(float);  // 640 MB
    if (ws_size >= kbytes) {
        float* Kd = (float*)d_ws;
        long long nk4 = (long long)INPUT_DIM * UNITS / 4;
        zero_f4_kernel<<<4096, 256, 0, stream>>>((float4*)Kd, nk4);
        zero_f4_kernel<<<1024, 256, 0, stream>>>((float4*)out, (long long)nout / 4);
        scatter_kernel<<<(nnz + 255) / 256, 256, 0, stream>>>(ind, val, Kd, nnz);
        dim3 grid(UNITS / 64, KSPLIT);  // 125 x 10
        gemm_splitk_kernel<<<grid, 256, 0, stream>>>(inputs, Kd, out);
        bias_tanh_kernel<<<(nout + 255) / 256, 256, 0, stream>>>(out, bias, nout);
    } else {
        zero_f4_kernel<<<1024, 256, 0, stream>>>((float4*)out, (long long)nout / 4);
        spmm_atomic_kernel<<<nnz, BATCH, 0, stream>>>(inputs, ind, val, out, nnz);
        bias_tanh_kernel<<<(nout + 255) / 256, 256, 0, stream>>>(out, bias, nout);
    }
}